// RelationModel_90692529422644
// MI455X (gfx1250) — compile-verified
//
#include <hip/hip_runtime.h>
#include <hip/hip_bf16.h>
#include <stdint.h>

#define N_NODE 20000
#define N_REL  1024
#define RR     (N_REL * N_REL)   // 1,048,576 cells per relation-relation matrix (2^20)

typedef float v2f __attribute__((ext_vector_type(2)));
typedef float v8f __attribute__((ext_vector_type(8)));

// ---------------------------------------------------------------------------
// Kernel 1: per-node head/tail degrees -> reciprocal (one wave32 per node row)
// ---------------------------------------------------------------------------
__global__ __launch_bounds__(128)
void degrees_kernel(const float* __restrict__ H, const float* __restrict__ T,
                    float* __restrict__ invDh, float* __restrict__ invDt) {
  int wave = (int)((blockIdx.x * blockDim.x + threadIdx.x) >> 5);
  int lane = threadIdx.x & 31;
  if (wave >= N_NODE) return;
  const float* hr = H + (size_t)wave * N_REL;
  const float* tr = T + (size_t)wave * N_REL;
  float sh = 0.f, st = 0.f;
  for (int r = lane; r < N_REL; r += 32) { sh += hr[r]; st += tr[r]; }
  #pragma unroll
  for (int off = 16; off > 0; off >>= 1) {
    sh += __shfl_down(sh, off, 32);
    st += __shfl_down(st, off, 32);
  }
  if (lane == 0) { invDh[wave] = 1.0f / sh; invDt[wave] = 1.0f / st; }
}

// ---------------------------------------------------------------------------
// Kernel 2: four Gram products via exact-fp32 WMMA 16x16x4.
// One wave owns a 32x32 output region (2x2 subtiles x 4 products = 128 VGPR
// of accumulators). Per 4-K chunk: 16 operand loads feed 16 WMMAs (1:1),
// halving L2 operand traffic vs a 16x16 tile. A-operand (Hn^T/Tn^T) and
// B-operand (H/T) share the same per-lane gather on row-major [n][col].
// ---------------------------------------------------------------------------
__device__ __forceinline__ void mma_step2(
    const float* __restrict__ H, const float* __restrict__ T,
    const float* __restrict__ invDh, const float* __restrict__ invDt,
    int n0, int half, int col, int r0, int s0,
    v8f (&acc_hh)[2][2], v8f (&acc_tt)[2][2],
    v8f (&acc_ht)[2][2], v8f (&acc_th)[2][2]) {
  const float4 wh4 = *(const float4*)(invDh + n0);   // wave-uniform broadcast
  const float4 wt4 = *(const float4*)(invDt + n0);
  float wh0 = half ? wh4.z : wh4.x;
  float wh1 = half ? wh4.w : wh4.y;
  float wt0 = half ? wt4.z : wt4.x;
  float wt1 = half ? wt4.w : wt4.y;
  size_t b0 = (size_t)(n0 + 2 * half) * N_REL;
  size_t b1 = b0 + N_REL;
  v2f ah[2], at[2], bh[2], bt[2];
  #pragma unroll
  for (int u = 0; u < 2; ++u) {
    int rc = r0 + 16 * u + col;
    int sc = s0 + 16 * u + col;
    float hr0 = H[b0 + rc], hr1 = H[b1 + rc];
    float tr0 = T[b0 + rc], tr1 = T[b1 + rc];
    float hs0 = H[b0 + sc], hs1 = H[b1 + sc];
    float ts0 = T[b0 + sc], ts1 = T[b1 + sc];
    ah[u] = (v2f){ hr0 * wh0, hr1 * wh1 };   // Hn^T subtile (16x4)
    at[u] = (v2f){ tr0 * wt0, tr1 * wt1 };   // Tn^T subtile
    bh[u] = (v2f){ hs0, hs1 };               // H subtile (4x16)
    bt[u] = (v2f){ ts0, ts1 };               // T subtile
  }
  #pragma unroll
  for (int i = 0; i < 2; ++i) {
    #pragma unroll
    for (int j = 0; j < 2; ++j) {
      acc_hh[i][j] = __builtin_amdgcn_wmma_f32_16x16x4_f32(false, ah[i], false, bh[j], (short)0, acc_hh[i][j], false, false);
      acc_tt[i][j] = __builtin_amdgcn_wmma_f32_16x16x4_f32(false, at[i], false, bt[j], (short)0, acc_tt[i][j], false, false);
      acc_ht[i][j] = __builtin_amdgcn_wmma_f32_16x16x4_f32(false, ah[i], false, bt[j], (short)0, acc_ht[i][j], false, false);
      acc_th[i][j] = __builtin_amdgcn_wmma_f32_16x16x4_f32(false, at[i], false, bh[j], (short)0, acc_th[i][j], false, false);
    }
  }
}

__global__ __launch_bounds__(128)
void relgemm_kernel(const float* __restrict__ H, const float* __restrict__ T,
                    const float* __restrict__ invDh, const float* __restrict__ invDt,
                    float* __restrict__ out) {
  int waveInBlk = threadIdx.x >> 5;
  int lane = threadIdx.x & 31;
  int tile = blockIdx.x * 4 + waveInBlk;   // 1024 tiles = 32x32 grid of 32x32 regions
  int r0 = (tile >> 5) * 32;
  int s0 = (tile & 31) * 32;
  int half = lane >> 4, col = lane & 15;
  v8f acc_hh[2][2] = {}; v8f acc_tt[2][2] = {};
  v8f acc_ht[2][2] = {}; v8f acc_th[2][2] = {};
  for (int n0 = 0; n0 < N_NODE; n0 += 4) {   // 5000 K-chunks, 16 WMMA each
    mma_step2(H, T, invDh, invDt, n0, half, col, r0, s0,
              acc_hh, acc_tt, acc_ht, acc_th);
  }
  #pragma unroll
  for (int i = 0; i < 2; ++i) {
    #pragma unroll
    for (int j = 0; j < 2; ++j) {
      #pragma unroll
      for (int e = 0; e < 8; ++e) {
        size_t o = (size_t)(r0 + 16 * i + e + 8 * half) * N_REL + (s0 + 16 * j + col);
        float vhh = acc_hh[i][j][e];
        float vtt = acc_tt[i][j][e];
        out[o]          = vhh;
        out[RR + o]     = vtt;
        out[2 * RR + o] = acc_ht[i][j][e];
        out[3 * RR + o] = acc_th[i][j][e];
        out[4 * RR + o] = vhh + vtt;   // A = Ahh + Att
      }
    }
  }
}

// ---------------------------------------------------------------------------
// Quantile-bin machinery: mask prefix scan, key build, bitonic sort, bins.
// ---------------------------------------------------------------------------
__global__ __launch_bounds__(256)
void scan_pass1(const float* __restrict__ adj, int* __restrict__ blockSums) {
  __shared__ int s[256];
  int t = threadIdx.x;
  size_t base = (size_t)blockIdx.x * 1024 + (size_t)t * 4;
  const float4 v = *(const float4*)(adj + base);
  int c = (v.x > 0.f) + (v.y > 0.f) + (v.z > 0.f) + (v.w > 0.f);
  s[t] = c; __syncthreads();
  for (int off = 1; off < 256; off <<= 1) {
    int u = (t >= off) ? s[t - off] : 0;
    __syncthreads();
    s[t] += u;
    __syncthreads();
  }
  if (t == 255) blockSums[blockIdx.x] = s[255];
}

__global__ void scan_blocksums(int* __restrict__ blockSums, int* __restrict__ nnz_out) {
  __shared__ int s[1024];
  int t = threadIdx.x;
  s[t] = blockSums[t]; __syncthreads();
  for (int off = 1; off < 1024; off <<= 1) {
    int u = (t >= off) ? s[t - off] : 0;
    __syncthreads();
    s[t] += u;
    __syncthreads();
  }
  blockSums[t] = (t == 0) ? 0 : s[t - 1];   // exclusive block offsets
  if (t == 1023) *nnz_out = s[1023];
}

__global__ __launch_bounds__(256)
void scan_pass2(const float* __restrict__ adj, const int* __restrict__ blockOff,
                int* __restrict__ cidx, unsigned long long* __restrict__ keys) {
  __shared__ int s[256];
  int t = threadIdx.x;
  size_t base = (size_t)blockIdx.x * 1024 + (size_t)t * 4;
  const float4 v = *(const float4*)(adj + base);
  float vals[4] = { v.x, v.y, v.z, v.w };
  int   ms[4]   = { v.x > 0.f, v.y > 0.f, v.z > 0.f, v.w > 0.f };
  int c = ms[0] + ms[1] + ms[2] + ms[3];
  s[t] = c; __syncthreads();
  for (int off = 1; off < 256; off <<= 1) {
    int u = (t >= off) ? s[t - off] : 0;
    __syncthreads();
    s[t] += u;
    __syncthreads();
  }
  int run = s[t] - c + blockOff[blockIdx.x];
  #pragma unroll
  for (int i = 0; i < 4; ++i) {
    size_t f = base + i;
    if (ms[i]) {
      cidx[f] = run;
      // positive floats: bit pattern is order-isomorphic; low 20 bits =
      // coalesced index reproduces jax's stable tie-break in flat order.
      unsigned vb = __float_as_uint(vals[i]);
      keys[f] = ((unsigned long long)vb << 20) | (unsigned)run;
      run++;
    } else {
      keys[f] = 0xFFFFFFFFFFFFFFFFull;   // padding -> sorts to the end
    }
  }
}

// Global compare-exchange step (only needed while j >= 2048).
__global__ __launch_bounds__(256)
void bitonic_step(unsigned long long* __restrict__ keys, int j, int k) {
  int i = (int)(blockIdx.x * blockDim.x + threadIdx.x);
  int ixj = i ^ j;
  if (ixj > i) {
    unsigned long long a = keys[i], b = keys[ixj];
    bool up = ((i & k) == 0);
    if ((a > b) == up) { keys[i] = b; keys[ixj] = a; }
  }
}

// All stages k=2..2048 on a 2048-key LDS block (direction uses GLOBAL index,
// so this is bit-identical to the global bitonic network).
__global__ __launch_bounds__(1024)
void bitonic_sort_local(unsigned long long* __restrict__ keys) {
  __shared__ unsigned long long s[2048];
  int t = threadIdx.x;
  int base = (int)blockIdx.x * 2048;
  s[t] = keys[base + t];
  s[t + 1024] = keys[base + t + 1024];
  __syncthreads();
  for (int k = 2; k <= 2048; k <<= 1) {
    for (int j = k >> 1; j > 0; j >>= 1) {
      #pragma unroll
      for (int w = 0; w < 2; ++w) {
        int i = t + w * 1024;
        int ixj = i ^ j;
        if (ixj > i) {
          unsigned long long a = s[i], b = s[ixj];
          bool up = (((base + i) & k) == 0);
          if ((a > b) == up) { s[i] = b; s[ixj] = a; }
        }
      }
      __syncthreads();
    }
  }
  keys[base + t] = s[t];
  keys[base + t + 1024] = s[t + 1024];
}

// Tail of stage k: all steps j=1024..1 fused in LDS.
__global__ __launch_bounds__(1024)
void bitonic_merge_local(unsigned long long* __restrict__ keys, int k) {
  __shared__ unsigned long long s[2048];
  int t = threadIdx.x;
  int base = (int)blockIdx.x * 2048;
  s[t] = keys[base + t];
  s[t + 1024] = keys[base + t + 1024];
  __syncthreads();
  for (int j = 1024; j > 0; j >>= 1) {
    #pragma unroll
    for (int w = 0; w < 2; ++w) {
      int i = t + w * 1024;
      int ixj = i ^ j;
      if (ixj > i) {
        unsigned long long a = s[i], b = s[ixj];
        bool up = (((base + i) & k) == 0);
        if ((a > b) == up) { s[i] = b; s[ixj] = a; }
      }
    }
    __syncthreads();
  }
  keys[base + t] = s[t];
  keys[base + t + 1024] = s[t + 1024];
}

// torch-faithful: bins_at_coalesced[p] = coalesced_idx(sorted[p]) * nb / nnz
__global__ __launch_bounds__(256)
void bac_kernel(const unsigned long long* __restrict__ keys,
                const int* __restrict__ nnz_ptr, const int* __restrict__ nb_ptr,
                int* __restrict__ bac) {
  int p = (int)(blockIdx.x * blockDim.x + threadIdx.x);
  int nnz = *nnz_ptr;
  if (p < nnz) {
    long long c = (long long)(keys[p] & 0xFFFFFu);
    bac[p] = (int)((c * (long long)(*nb_ptr)) / (long long)nnz);
  }
}

__global__ __launch_bounds__(256)
void scatter_bins(const float* __restrict__ adj, const int* __restrict__ cidx,
                  const int* __restrict__ bac, float* __restrict__ bins_out) {
  size_t f = (size_t)blockIdx.x * blockDim.x + threadIdx.x;
  float v = adj[f];
  bins_out[f] = (v > 0.f) ? (float)bac[cidx[f]] : -1.0f;
}

// ---------------------------------------------------------------------------
extern "C" void kernel_launch(void* const* d_in, const int* in_sizes, int n_in,
                              void* d_out, int out_size, void* d_ws, size_t ws_size,
                              hipStream_t stream) {
  const float* H  = (const float*)d_in[0];
  const float* T  = (const float*)d_in[1];
  const int*   nb = (const int*)d_in[2];
  float* out = (float*)d_out;                 // [5*RR adj][5*RR bins-as-float]
  char*  ws  = (char*)d_ws;

  float* invDh = (float*)ws;                              // 20000 f32
  float* invDt = invDh + 20480;                           // 20000 f32
  unsigned long long* keys = (unsigned long long*)(ws + (size_t)(1 << 18)); // 8 MB
  int* cidx      = (int*)(ws + (size_t)(1 << 18) + (size_t)8 * RR);          // 4 MB
  int* bac       = cidx + RR;                                                // 4 MB
  int* blockSums = bac + RR;                                                 // 1024
  int* nnz       = blockSums + 1024;

  // 1) degrees (20000 waves)
  degrees_kernel<<<5000, 128, 0, stream>>>(H, T, invDh, invDt);

  // 2) the four Gram products + sum, on the WMMA pipe (1024 waves, 32x32 each)
  relgemm_kernel<<<256, 128, 0, stream>>>(H, T, invDh, invDt, out);

  // 3) per-matrix quantile bins
  for (int m = 0; m < 5; ++m) {
    const float* adjm = out + (size_t)m * RR;
    float* binsm = out + (size_t)5 * RR + (size_t)m * RR;

    scan_pass1<<<1024, 256, 0, stream>>>(adjm, blockSums);
    scan_blocksums<<<1, 1024, 0, stream>>>(blockSums, nnz);
    scan_pass2<<<1024, 256, 0, stream>>>(adjm, blockSums, cidx, keys);

    // LDS-fused bitonic sort: stages k<=2048 in one kernel, then for each
    // larger k only the j>=2048 steps run globally, tail fused in LDS.
    bitonic_sort_local<<<RR / 2048, 1024, 0, stream>>>(keys);
    for (int k = 4096; k <= RR; k <<= 1) {
      for (int j = k >> 1; j >= 2048; j >>= 1)
        bitonic_step<<<RR / 256, 256, 0, stream>>>(keys, j, k);
      bitonic_merge_local<<<RR / 2048, 1024, 0, stream>>>(keys, k);
    }

    bac_kernel<<<RR / 256, 256, 0, stream>>>(keys, nnz, nb, bac);
    scatter_bins<<<RR / 256, 256, 0, stream>>>(adjm, cidx, bac, binsm);
  }
}